// SAConvLstmCell_10934986735699
// MI455X (gfx1250) — compile-verified
//
#include <hip/hip_runtime.h>
#include <hip/hip_bf16.h>
#include <math.h>
#include <stdint.h>

// ---------------------------------------------------------------------------
// SA-ConvLSTM cell for MI455X (gfx1250, wave32).  All GEMM-shaped work runs
// through V_WMMA_F32_16X16X4_F32 (fp32 in/out matches reference dtype).
// Conv weight-tile staging uses GLOBAL_LOAD_ASYNC_TO_LDS (ASYNCcnt) when the
// toolchain exposes the builtins, with a VGPR+ds_store fallback otherwise.
// ---------------------------------------------------------------------------

typedef __attribute__((ext_vector_type(2))) float v2f;
typedef __attribute__((ext_vector_type(8))) float v8f;

#define NPIX 1024   // H*W
#define BATCH 16
#define CCH 64      // CIN == COUT
#define HID 32

#define AS1 __attribute__((address_space(1)))
#define AS3 __attribute__((address_space(3)))

#if __has_builtin(__builtin_amdgcn_global_load_async_to_lds_b32) && \
    __has_builtin(__builtin_amdgcn_s_wait_asynccnt)
#define HAVE_ASYNC_LDS 1
#else
#define HAVE_ASYNC_LDS 0
#endif

__device__ __forceinline__ v8f wmma4(v2f a, v2f b, v8f c) {
  // (neg_a, A, neg_b, B, c_mod, C, reuse_a, reuse_b)
  return __builtin_amdgcn_wmma_f32_16x16x4_f32(false, a, false, b, (short)0, c,
                                               false, false);
}

__device__ __forceinline__ float sigmoidf_(float x) {
  return 1.0f / (1.0f + __expf(-x));
}

// ---------------------------------------------------------------------------
// Kernel 1: channel projections.  q = Wq(32x64) @ xf(64xN), k likewise,
// v stored TRANSPOSED: vT[n][c] = (Wv @ xf)[c][n]  (coalesced B-operand later)
// grid: (B, 16) x 128 threads (4 waves, each owns 16 columns of n)
// ---------------------------------------------------------------------------
__global__ __launch_bounds__(128) void proj_kernel(
    const float* __restrict__ xf,   // [B, 64, N]
    const float* __restrict__ Wq,   // [32, 64]
    const float* __restrict__ Wk,   // [32, 64]
    const float* __restrict__ Wv,   // [64, 64]
    float* __restrict__ qb,         // [B, 32, N]
    float* __restrict__ kb,         // [B, 32, N]
    float* __restrict__ vT)         // [B, N, 64]
{
  const int b    = blockIdx.x;
  const int grp  = blockIdx.y;            // 64-column chunk
  const int wave = threadIdx.x >> 5;
  const int lane = threadIdx.x & 31;
  const int half = lane >> 4;
  const int l16  = lane & 15;
  const int n0   = grp * 64 + wave * 16;

  const float* xb = xf + b * CCH * NPIX;

  v8f accq0 = {}, accq1 = {}, acck0 = {}, acck1 = {};
  v8f accv0 = {}, accv1 = {}, accv2 = {}, accv3 = {};

  for (int c0 = 0; c0 < CCH; c0 += 4) {
    const int kc = c0 + 2 * half;          // K index pair for this lane-half
    v2f bx;                                 // B operand: x[kc..kc+1][n0+l16]
    bx.x = xb[(kc + 0) * NPIX + n0 + l16];
    bx.y = xb[(kc + 1) * NPIX + n0 + l16];

    v2f a;
    a.x = Wq[(l16)      * CCH + kc]; a.y = Wq[(l16)      * CCH + kc + 1];
    accq0 = wmma4(a, bx, accq0);
    a.x = Wq[(16 + l16) * CCH + kc]; a.y = Wq[(16 + l16) * CCH + kc + 1];
    accq1 = wmma4(a, bx, accq1);
    a.x = Wk[(l16)      * CCH + kc]; a.y = Wk[(l16)      * CCH + kc + 1];
    acck0 = wmma4(a, bx, acck0);
    a.x = Wk[(16 + l16) * CCH + kc]; a.y = Wk[(16 + l16) * CCH + kc + 1];
    acck1 = wmma4(a, bx, acck1);

    a.x = Wv[(0 * 16 + l16) * CCH + kc]; a.y = Wv[(0 * 16 + l16) * CCH + kc + 1];
    accv0 = wmma4(a, bx, accv0);
    a.x = Wv[(1 * 16 + l16) * CCH + kc]; a.y = Wv[(1 * 16 + l16) * CCH + kc + 1];
    accv1 = wmma4(a, bx, accv1);
    a.x = Wv[(2 * 16 + l16) * CCH + kc]; a.y = Wv[(2 * 16 + l16) * CCH + kc + 1];
    accv2 = wmma4(a, bx, accv2);
    a.x = Wv[(3 * 16 + l16) * CCH + kc]; a.y = Wv[(3 * 16 + l16) * CCH + kc + 1];
    accv3 = wmma4(a, bx, accv3);
  }

  float* qo = qb + b * HID * NPIX;
  float* ko = kb + b * HID * NPIX;
  float* vo = vT + b * NPIX * CCH;
#pragma unroll
  for (int j = 0; j < 8; ++j) {
    const int d = j + 8 * half;            // D row (M index)
    qo[(d)      * NPIX + n0 + l16] = accq0[j];
    qo[(16 + d) * NPIX + n0 + l16] = accq1[j];
    ko[(d)      * NPIX + n0 + l16] = acck0[j];
    ko[(16 + d) * NPIX + n0 + l16] = acck1[j];
    vo[(n0 + l16) * CCH + (0 * 16 + d)] = accv0[j];
    vo[(n0 + l16) * CCH + (1 * 16 + d)] = accv1[j];
    vo[(n0 + l16) * CCH + (2 * 16 + d)] = accv2[j];
    vo[(n0 + l16) * CCH + (3 * 16 + d)] = accv3[j];
  }
}

// ---------------------------------------------------------------------------
// Kernel 2: fused attention core for a 16-row block of scores.
//   scores[16,1024] = q^T k  (WMMA, stored to padded LDS)
//   softmax rows in LDS (optionally emit `atten` rows)
//   out[n,c] = attn @ v^T   (WMMA, A from LDS)
//   result[c,n] = scale * (resid[c,n] + out[n,c])
// grid: (B, 64) x 128 threads; LDS: 16 rows, stride 1026 (bank-conflict-free)
// ---------------------------------------------------------------------------
#define SROW 1026
__global__ __launch_bounds__(128) void attn_kernel(
    const float* __restrict__ qb,    // [B,32,N]
    const float* __restrict__ kb,    // [B,32,N]
    const float* __restrict__ vT,    // [B,N,64]
    const float* __restrict__ resid, // [B,64,N] with batch stride rs
    float* __restrict__ outp,        // [*,64,N] with batch stride os
    float* __restrict__ atten,       // [B,N,N]
    int rs, int os, float scale, int write_atten)
{
  __shared__ float sc[16 * SROW];
  __shared__ float red[128];

  const int b    = blockIdx.x;
  const int nb   = blockIdx.y;
  const int n0   = nb * 16;
  const int wave = threadIdx.x >> 5;
  const int lane = threadIdx.x & 31;
  const int half = lane >> 4;
  const int l16  = lane & 15;

  const float* q = qb + b * HID * NPIX;
  const float* k = kb + b * HID * NPIX;

  // ---- phase 1: 16x1024 scores block, 16 column-tiles per wave -----------
  for (int mt = wave; mt < 64; mt += 4) {
    const int m0 = mt * 16;
    v8f acc = {};
#pragma unroll
    for (int d0 = 0; d0 < HID; d0 += 4) {
      const int kd = d0 + 2 * half;
      v2f a, bbv;
      a.x   = q[(kd)     * NPIX + n0 + l16];
      a.y   = q[(kd + 1) * NPIX + n0 + l16];
      bbv.x = k[(kd)     * NPIX + m0 + l16];
      bbv.y = k[(kd + 1) * NPIX + m0 + l16];
      acc = wmma4(a, bbv, acc);
    }
#pragma unroll
    for (int j = 0; j < 8; ++j)
      sc[(j + 8 * half) * SROW + m0 + l16] = acc[j];
  }
  __syncthreads();

  // ---- phase 2: softmax over each of the 16 rows (8 threads / row) -------
  const int r = threadIdx.x >> 3;
  const int t = threadIdx.x & 7;
  float mx = -INFINITY;
  for (int m = t; m < NPIX; m += 8) mx = fmaxf(mx, sc[r * SROW + m]);
  red[threadIdx.x] = mx;
  __syncthreads();
#pragma unroll
  for (int i = 0; i < 8; ++i) mx = fmaxf(mx, red[r * 8 + i]);

  float sum = 0.0f;
  for (int m = t; m < NPIX; m += 8) {
    const float e = __expf(sc[r * SROW + m] - mx);
    sc[r * SROW + m] = e;
    sum += e;
  }
  __syncthreads();            // red[] reuse hazard
  red[threadIdx.x] = sum;
  __syncthreads();
  sum = 0.0f;
#pragma unroll
  for (int i = 0; i < 8; ++i) sum += red[r * 8 + i];
  const float inv = 1.0f / sum;

  float* arow = atten + (size_t)b * NPIX * NPIX + (size_t)(n0 + r) * NPIX;
  for (int m = t; m < NPIX; m += 8) {
    const float p = sc[r * SROW + m] * inv;
    sc[r * SROW + m] = p;
    if (write_atten) arow[m] = p;
  }
  __syncthreads();

  // ---- phase 3: out = attn @ v^T ; one 16-channel tile per wave ----------
  const int c0 = wave * 16;
  const float* vt = vT + b * NPIX * CCH;
  v8f acc = {};
  for (int m0 = 0; m0 < NPIX; m0 += 4) {
    const int km = m0 + 2 * half;
    v2f a, bbv;
    a.x   = sc[l16 * SROW + km];
    a.y   = sc[l16 * SROW + km + 1];
    bbv.x = vt[(km)     * CCH + c0 + l16];
    bbv.y = vt[(km + 1) * CCH + c0 + l16];
    acc = wmma4(a, bbv, acc);
  }
  const float* rb = resid + (size_t)b * rs;
  float*       ob = outp  + (size_t)b * os;
#pragma unroll
  for (int j = 0; j < 8; ++j) {
    const int n = n0 + j + 8 * half;
    const int c = c0 + l16;
    ob[c * NPIX + n] = scale * (rb[c * NPIX + n] + acc[j]);
  }
}

// ---------------------------------------------------------------------------
// Kernel 3: copy h into second half of combined buffer [B,128,32,32]
// ---------------------------------------------------------------------------
__global__ void copyh_kernel(const float* __restrict__ h, float* __restrict__ comb)
{
  const int idx = blockIdx.x * blockDim.x + threadIdx.x;   // B*64*1024
  if (idx >= BATCH * CCH * NPIX) return;
  const int b = idx >> 16;          // 64*1024
  const int rem = idx & 65535;
  comb[b * 128 * NPIX + CCH * NPIX + rem] = h[idx];
}

// ---------------------------------------------------------------------------
// Kernel 4: 5x5 conv as implicit GEMM.  M=16 out-channels (LDS weight tile per
// tap, staged with GLOBAL_LOAD_ASYNC_TO_LDS when available), N=16 pixels,
// K = 128 in-channels x 25 taps.
// grid: (B, 16 oc-tiles, 16 pixel-chunks) x 128 threads
// ---------------------------------------------------------------------------
#define WROW 130
__global__ __launch_bounds__(128) void conv_kernel(
    const float* __restrict__ comb,  // [B,128,32,32]
    const float* __restrict__ W,     // [256,128,5,5]
    const float* __restrict__ bias,  // [256]
    float* __restrict__ gates)       // [B,256,1024]
{
  __shared__ float sw[16 * WROW];    // [oc 16][ic 128], padded stride

  const int b    = blockIdx.x;
  const int ot   = blockIdx.y;                 // oc tile
  const int pc   = blockIdx.z;                 // pixel chunk
  const int wave = threadIdx.x >> 5;
  const int lane = threadIdx.x & 31;
  const int half = lane >> 4;
  const int l16  = lane & 15;
  const int p0   = pc * 64 + wave * 16;        // 16 pixels, single image row
  const int py   = p0 >> 5;
  const int px0  = p0 & 31;

  const float* cbase = comb + (size_t)b * 128 * NPIX;
  v8f acc = {};

  for (int tap = 0; tap < 25; ++tap) {
    const int ky = tap / 5, kx = tap % 5;
    __syncthreads();                           // protect previous tile use
#if HAVE_ASYNC_LDS
    // Stage the 16x128 weight tile for this tap straight into LDS via the
    // async copy unit (ASYNCcnt); no VGPR round-trip, frees VALU for WMMA.
    for (int i = threadIdx.x; i < 16 * 128; i += 128) {
      const int oc = i >> 7, ic = i & 127;
      __builtin_amdgcn_global_load_async_to_lds_b32(
          (AS1 int*)(uintptr_t)&W[(size_t)(ot * 16 + oc) * 3200 + ic * 25 + tap],
          (AS3 int*)(uintptr_t)&sw[oc * WROW + ic], 0, 0);
    }
    __builtin_amdgcn_s_wait_asynccnt(0);       // this wave's copies landed
    __syncthreads();                           // all waves' copies landed
#else
    for (int i = threadIdx.x; i < 16 * 128; i += 128) {
      const int oc = i >> 7, ic = i & 127;
      sw[oc * WROW + ic] = W[(size_t)(ot * 16 + oc) * 3200 + ic * 25 + tap];
      if (tap < 24)                            // warm next tap's weights
        __builtin_prefetch(&W[(size_t)(ot * 16 + oc) * 3200 + ic * 25 + tap + 1], 0, 1);
    }
    __syncthreads();
#endif

    const int y  = py + ky - 2;
    const int xx = px0 + l16 + kx - 2;
    const bool valid = (y >= 0) && (y < 32) && (xx >= 0) && (xx < 32);
    const int pix = valid ? (y * 32 + xx) : 0; // clamped safe address
    const float* cb = cbase + pix;

    for (int ic0 = 0; ic0 < 128; ic0 += 4) {
      const int kc = ic0 + 2 * half;
      v2f a, bbv;
      a.x = sw[l16 * WROW + kc];
      a.y = sw[l16 * WROW + kc + 1];
      const float v0 = cb[(kc)     * NPIX];
      const float v1 = cb[(kc + 1) * NPIX];
      bbv.x = valid ? v0 : 0.0f;
      bbv.y = valid ? v1 : 0.0f;
      acc = wmma4(a, bbv, acc);
    }
  }

  float* go = gates + (size_t)b * 256 * NPIX;
#pragma unroll
  for (int j = 0; j < 8; ++j) {
    const int oc = ot * 16 + j + 8 * half;
    go[oc * NPIX + p0 + l16] = acc[j] + bias[oc];
  }
}

// ---------------------------------------------------------------------------
// Kernel 5: LSTM gate nonlinearities + state update.
// ---------------------------------------------------------------------------
__global__ void lstm_kernel(const float* __restrict__ gates,
                            const float* __restrict__ c_in,
                            float* __restrict__ c_out,
                            float* __restrict__ h_out)
{
  const int idx = blockIdx.x * blockDim.x + threadIdx.x;   // B*64*1024
  if (idx >= BATCH * CCH * NPIX) return;
  const int b = idx >> 16;
  const int rem = idx & 65535;
  const float* gb = gates + (size_t)b * 256 * NPIX;
  const float i = sigmoidf_(gb[0 * 65536 + rem]);
  const float f = sigmoidf_(gb[1 * 65536 + rem]);
  const float o = sigmoidf_(gb[2 * 65536 + rem]);
  const float g = tanhf(gb[3 * 65536 + rem]);
  const float cn = f * c_in[idx] + i * g;
  c_out[idx] = cn;
  h_out[idx] = o * tanhf(cn);
}

// ---------------------------------------------------------------------------
extern "C" void kernel_launch(void* const* d_in, const int* in_sizes, int n_in,
                              void* d_out, int out_size, void* d_ws, size_t ws_size,
                              hipStream_t stream) {
  (void)in_sizes; (void)n_in; (void)out_size; (void)ws_size;
  const float* x     = (const float*)d_in[0];
  const float* h     = (const float*)d_in[1];
  const float* c     = (const float*)d_in[2];
  const float* Wq_x  = (const float*)d_in[3];
  const float* Wk_x  = (const float*)d_in[4];
  const float* Wv_x  = (const float*)d_in[5];
  const float* Wq_h  = (const float*)d_in[6];
  const float* Wk_h  = (const float*)d_in[7];
  const float* Wv_h  = (const float*)d_in[8];
  const float* Wlstm = (const float*)d_in[9];
  const float* blstm = (const float*)d_in[10];

  float* out   = (float*)d_out;
  float* newh  = out;                        // [16,64,32,32]
  float* newc  = out + 1048576;              // [16,64,32,32]
  float* atten = out + 2097152;              // [16,1024,1024]

  float* ws    = (float*)d_ws;
  float* comb  = ws;                         // [16,128,1024]  (X in ch 0..63)
  float* qb    = comb + 2097152;             // [16,32,1024]
  float* kb    = qb   + 524288;
  float* vT    = kb   + 524288;              // [16,1024,64]
  float* gates = vT   + 1048576;             // [16,256,1024]
  float* hnew  = gates + 4194304;            // [16,64,1024]

  // ---- attention #1 over x, residual into comb channels [0,64) ----------
  proj_kernel<<<dim3(BATCH, 16), 128, 0, stream>>>(x, Wq_x, Wk_x, Wv_x, qb, kb, vT);
  attn_kernel<<<dim3(BATCH, 64), 128, 0, stream>>>(
      qb, kb, vT, x, comb, atten,
      /*rs=*/CCH * NPIX, /*os=*/128 * NPIX, /*scale=*/1.0f, /*write_atten=*/0);

  // ---- ConvLSTM ----------------------------------------------------------
  copyh_kernel<<<4096, 256, 0, stream>>>(h, comb);
  conv_kernel<<<dim3(BATCH, 16, 16), 128, 0, stream>>>(comb, Wlstm, blstm, gates);
  lstm_kernel<<<4096, 256, 0, stream>>>(gates, c, newc, hnew);

  // ---- attention #2 over new_h: writes atten and 2*(h_new + attn_out) ----
  proj_kernel<<<dim3(BATCH, 16), 128, 0, stream>>>(hnew, Wq_h, Wk_h, Wv_h, qb, kb, vT);
  attn_kernel<<<dim3(BATCH, 64), 128, 0, stream>>>(
      qb, kb, vT, hnew, newh, atten,
      /*rs=*/CCH * NPIX, /*os=*/CCH * NPIX, /*scale=*/2.0f, /*write_atten=*/1);
}